// CauDilConv1D_62242666053644
// MI455X (gfx1250) — compile-verified
//
#include <hip/hip_runtime.h>

// Causal dilated conv1d as fp32 WMMA GEMM for gfx1250 (MI455X).
// out[b,co,t] = sum_{ci,kw} W[co, ci*3+kw] * x[b,ci, t + 4*kw - 8]
// M=32 (co), K=96 (reordered as kw*32+ci), N=L (time), per batch.
// HBM-bound: ~512 MB traffic -> ~23 us floor at 23.3 TB/s. Design goal:
// minimize dynamic instructions/byte so issue rate stays under that floor.

typedef __attribute__((ext_vector_type(2))) float v2f;
typedef __attribute__((ext_vector_type(8))) float v8f;

#define CD_C_IN   32
#define CD_C_OUT  32
#define CD_KW     3
#define CD_DIL    4
#define CD_PAD    8
#define CD_L      131072
#define CD_KTOT   96                     // C_IN * KW
#define CD_NT     8                      // 16-wide time tiles per wave
#define CD_WAVES  8
#define CD_T_PER_WAVE  (16 * CD_NT)      // 128
#define CD_T_PER_BLOCK (CD_T_PER_WAVE * CD_WAVES)  // 1024

template <bool GUARD>
__device__ __forceinline__ void cd_conv_wave(
    const float* __restrict__ xb,    // x for this batch: [C_IN][L]
    const float* __restrict__ Wm,    // weights: [C_OUT][96] row-major
    float* __restrict__ outb,        // out for this batch: [C_OUT][L]
    int co_base, int t0)
{
    const int lane = (int)(threadIdx.x & 31u);
    const int nlo  = lane & 15;      // A row (co) / B,C column (t)
    const int hi   = lane >> 4;      // 0 or 1

    v8f acc[CD_NT] = {};             // zero-init accumulators

    // Per-lane base pointers; every per-kk / per-nt delta below is a
    // compile-time constant so loads/stores use immediate offsets.
    const float* wr = Wm + (co_base + nlo) * CD_KTOT + 6 * hi;
    const float* xw = xb + (size_t)(2 * hi) * CD_L + (uint32_t)(t0 + nlo);

    #pragma unroll
    for (int kk = 0; kk < 24; ++kk) {
        const int kw     = kk >> 3;                 // 0,1,2 (uniform)
        const int toff   = kw * CD_DIL - CD_PAD;    // -8,-4,0
        const int cconst = (kk * 4) & 31;           // ci base (v=0,hi=0)

        // A fragment: lane l, vgpr v -> W[co_base+l%16, (cconst+2hi+v)*3+kw]
        v2f a;
        a.x = wr[cconst * CD_KW + kw];              // imm offsets <= 380 B
        a.y = wr[cconst * CD_KW + kw + CD_KW];

        // One constant 64-bit add per kk; all 16 loads below are imm-offset.
        const float* xr = xw + (ptrdiff_t)(cconst * CD_L + toff);

        #pragma unroll
        for (int nt = 0; nt < CD_NT; ++nt) {
            v2f b;
            if constexpr (GUARD) {
                const int t = t0 + nlo + nt * 16 + toff;   // t0 == 0 here
                b.x = (t >= 0) ? xr[nt * 16] : 0.0f;
                b.y = (t >= 0) ? xr[CD_L + nt * 16] : 0.0f;
            } else {
                b.x = xr[nt * 16];            // ci0 row   (imm: nt*64 B)
                b.y = xr[CD_L + nt * 16];     // ci0+1 row (imm: +512 KB)
            }
            // D = A(16x4) * B(4x16) + C  -> v_wmma_f32_16x16x4_f32
            acc[nt] = __builtin_amdgcn_wmma_f32_16x16x4_f32(
                false, a, false, b, (short)0, acc[nt], false, false);
        }
    }

    // Store: C/D layout vgpr v -> row co_base + v + 8*hi, col t0+nt*16+nlo.
    // All 64 stores are imm offsets (nt*64 B + v*512 KB <= 3.67 MB) off one
    // pointer. Nontemporal: output never re-read, keep L2 for x overlap.
    float* orow = outb + (size_t)(co_base + 8 * hi) * CD_L
                       + (uint32_t)(t0 + nlo);
    #pragma unroll
    for (int nt = 0; nt < CD_NT; ++nt) {
        #pragma unroll
        for (int v = 0; v < 8; ++v) {
            __builtin_nontemporal_store(acc[nt][v],
                                        orow + nt * 16 + (size_t)v * CD_L);
        }
    }
}

__global__ void __launch_bounds__(256)
caudilconv1d_wmma_f32(const float* __restrict__ x,
                      const float* __restrict__ Wm,
                      float* __restrict__ out)
{
    const int wave    = (int)(threadIdx.x >> 5);
    const int t0      = (int)blockIdx.x * CD_T_PER_BLOCK + wave * CD_T_PER_WAVE;
    const int co_base = (int)blockIdx.y * 16;
    const int b       = (int)blockIdx.z;

    const float* xb   = x   + (size_t)b * CD_C_IN  * CD_L;
    float*       outb = out + (size_t)b * CD_C_OUT * CD_L;

    if (t0 == 0) {
        cd_conv_wave<true >(xb, Wm, outb, co_base, t0);   // t-8 can underflow
    } else {
        cd_conv_wave<false>(xb, Wm, outb, co_base, t0);   // interior fast path
    }
}

extern "C" void kernel_launch(void* const* d_in, const int* in_sizes, int n_in,
                              void* d_out, int out_size, void* d_ws, size_t ws_size,
                              hipStream_t stream)
{
    const float* x  = (const float*)d_in[0];   // (16, 32, 131072) fp32
    const float* Wm = (const float*)d_in[1];   // (32, 96) fp32
    float* out      = (float*)d_out;           // (16, 32, 131072) fp32

    dim3 grid(CD_L / CD_T_PER_BLOCK,           // 128 time super-tiles
              CD_C_OUT / 16,                   // 2 co halves
              16);                             // batch
    caudilconv1d_wmma_f32<<<grid, 256, 0, stream>>>(x, Wm, out);
}